// MoeFcTokensConvolution_31275951850273
// MI455X (gfx1250) — compile-verified
//
#include <hip/hip_runtime.h>
#include <math.h>

// ---------------- problem constants (match reference) ----------------
constexpr int B = 32, S = 2048, D = 1024, E = 16, KTOP = 4, O = 1024;

typedef float v2f __attribute__((ext_vector_type(2)));
typedef float v8f __attribute__((ext_vector_type(8)));
typedef int   v4i __attribute__((vector_size(16)));   // matches builtin param type

// D = A(16x4 f32) * B(4x16 f32) + C(16x16 f32)   -> v_wmma_f32_16x16x4_f32
#define WMMA_F32_4(a, b, c) \
  __builtin_amdgcn_wmma_f32_16x16x4_f32(false, (a), false, (b), (short)0, (c), false, false)

// CDNA5 async global->LDS path (ASYNCcnt-tracked, bypasses VGPRs)
#if defined(__HIP_DEVICE_COMPILE__) && defined(__has_builtin)
#if __has_builtin(__builtin_amdgcn_global_load_async_to_lds_b128)
#define HAVE_ASYNC_LDS 1
#endif
#if __has_builtin(__builtin_amdgcn_s_wait_asynccnt)
#define HAVE_WAIT_ASYNC 1
#endif
#endif

#if defined(HAVE_ASYNC_LDS)
// global (addrspace 1) int4* -> LDS (addrspace 3) int4*, 16 bytes per lane
#define GASYNC_B128(gp, lp)                                     \
  __builtin_amdgcn_global_load_async_to_lds_b128(               \
      (__attribute__((address_space(1))) v4i*)(gp),             \
      (__attribute__((address_space(3))) v4i*)(lp), 0, 0)
#endif

__device__ __forceinline__ void wait_async_zero() {
#if defined(HAVE_WAIT_ASYNC)
  __builtin_amdgcn_s_wait_asynccnt(0);
#elif defined(HAVE_ASYNC_LDS)
  asm volatile("s_wait_asynccnt 0x0" ::: "memory");
#endif
}

// =====================================================================
// Kernel 1: gate logits  logits[b*S+s][e] = x[b,s,:].Wg[:,e] + bg[e]
// 8 waves/block, 16 tokens per wave, Wg staged in LDS in K-chunks of 128.
// =====================================================================
__global__ __launch_bounds__(256) void gate_kernel(
    const float* __restrict__ x, const float* __restrict__ Wg,
    const float* __restrict__ bg, float* __restrict__ logits) {
  __shared__ __align__(16) float wg_lds[128 * E];  // 8 KB
  const int lane = threadIdx.x & 31;
  const int wave = threadIdx.x >> 5;
  const int tok0 = blockIdx.x * 128 + wave * 16;
  const int mrow = lane & 15;                 // A-fragment row (token within tile)
  const int koff = (lane < 16) ? 0 : 2;       // A/B fragment K offset
  const float* xrow = x + (size_t)(tok0 + mrow) * D;

  v8f acc = {0.f, 0.f, 0.f, 0.f, 0.f, 0.f, 0.f, 0.f};
  for (int k0 = 0; k0 < D; k0 += 128) {
    __syncthreads();
    {  // stage Wg[k0..k0+127][0..15]: 2048 floats = 512 float4, 2 per thread
      const float4* src = (const float4*)(Wg + (size_t)k0 * E);
      float4* dst = (float4*)wg_lds;
      dst[threadIdx.x] = src[threadIdx.x];
      dst[threadIdx.x + 256] = src[threadIdx.x + 256];
    }
    __syncthreads();
#pragma unroll
    for (int kk = 0; kk < 128; kk += 4) {
      const float2 av = *(const float2*)(xrow + k0 + kk + koff);
      v2f a;  a.x = av.x;  a.y = av.y;
      v2f b;
      b.x = wg_lds[(kk + koff) * E + mrow];       // B[k][n], n = lane%16
      b.y = wg_lds[(kk + koff + 1) * E + mrow];
      acc = WMMA_F32_4(a, b, acc);
    }
  }
  // C/D layout: vgpr r -> m = r + (lane<16 ? 0 : 8), n = lane%16
  const int n = lane & 15;
  const float bias = bg[n];
  const int mbase = (lane < 16) ? 0 : 8;
#pragma unroll
  for (int r = 0; r < 8; ++r) {
    const int tok = tok0 + mbase + r;
    logits[(size_t)tok * E + n] = acc[r] + bias;
  }
}

// =====================================================================
// Kernel 2: per (b,e): softmax stats over S tokens + top-4 (val,idx).
// Top-k of softmax == top-k of logits (softmax is monotone per column).
// =====================================================================
__global__ __launch_bounds__(256) void topk_kernel(
    const float* __restrict__ logits, int* __restrict__ topk_idx,
    float* __restrict__ topk_val) {
  const int be = blockIdx.x;
  const int b = be / E, e = be % E;
  __shared__ float v_lds[S];     // 8 KB
  __shared__ float red_v[256];
  __shared__ int   red_i[256];
  const int t = threadIdx.x;

  for (int s = t; s < S; s += 256)
    v_lds[s] = logits[((size_t)b * S + s) * E + e];
  __syncthreads();

  float topv[KTOP];
  int topi[KTOP];
  for (int r = 0; r < KTOP; ++r) {   // 4 argmax rounds (round 0 == global max)
    float bv = -__builtin_inff();
    int bi = 0x7fffffff;
    for (int s = t; s < S; s += 256) {
      const float v = v_lds[s];
      if (v > bv || (v == bv && s < bi)) { bv = v; bi = s; }
    }
    red_v[t] = bv;  red_i[t] = bi;
    __syncthreads();
    for (int off = 128; off > 0; off >>= 1) {
      if (t < off) {
        const float ov = red_v[t + off];
        const int oi = red_i[t + off];
        if (ov > red_v[t] || (ov == red_v[t] && oi < red_i[t])) {
          red_v[t] = ov;  red_i[t] = oi;
        }
      }
      __syncthreads();
    }
    topv[r] = red_v[0];  topi[r] = red_i[0];
    if (t == 0) v_lds[topi[r]] = -__builtin_inff();
    __syncthreads();
  }
  const float gmax = topv[0];
  // sum of exp over all tokens: masked slots were top-k, add them back
  float part = 0.f;
  for (int s = t; s < S; s += 256) {
    const float v = v_lds[s];
    if (v > -__builtin_inff()) part += __expf(v - gmax);
  }
  red_v[t] = part;
  __syncthreads();
  for (int off = 128; off > 0; off >>= 1) {
    if (t < off) red_v[t] += red_v[t + off];
    __syncthreads();
  }
  float sum = red_v[0];
#pragma unroll
  for (int r = 0; r < KTOP; ++r) sum += __expf(topv[r] - gmax);

  if (t < KTOP) {
    topk_idx[be * KTOP + t] = topi[t];
    topk_val[be * KTOP + t] = __expf(topv[t] - gmax) / sum;
  }
}

// =====================================================================
// Kernel 3: gather+scale selected tokens: inp[b,e, j*D+d] = x[b,idx_j,d]*val_j
// =====================================================================
__global__ __launch_bounds__(256) void gather_kernel(
    const float* __restrict__ x, const int* __restrict__ topk_idx,
    const float* __restrict__ topk_val, float* __restrict__ inp) {
  const int be = blockIdx.x;
  const int b = be / E;
#pragma unroll
  for (int j = 0; j < KTOP; ++j) {
    const int s = topk_idx[be * KTOP + j];
    const float v = topk_val[be * KTOP + j];
    const float* src = x + ((size_t)b * S + s) * D;
    float* dst = inp + (size_t)be * (KTOP * D) + (size_t)j * D;
    for (int d = threadIdx.x; d < D; d += 256) dst[d] = src[d] * v;
  }
}

// =====================================================================
// Kernels 4-6: grouped per-expert GEMM with fused bias(+relu).
//   C[m, e, n] = act( sum_k A[m, e, k] * W[e, k, n] + bias[e, n] )
// M = 32 rows (batch), A row stride = E*Kdim, C row stride = E*N.
// Block: 128 thr = 4 waves, 64-wide N strip, K chunked by 32.
// Double-buffered LDS; staging via CDNA5 async global->LDS (ASYNCcnt).
// =====================================================================
constexpr int BN = 64;   // N per block
constexpr int KC = 32;   // K chunk
constexpr int LDA = 36;  // padded LDS stride (conflict-free A fragments)

// stage one K-chunk: W[k0..k0+KC)[nbase..nbase+BN) and A[0..31][k0..k0+KC)
__device__ __forceinline__ void stage_chunk(
    const float* __restrict__ Wb, const float* __restrict__ Ab,
    float* __restrict__ wbuf, float* __restrict__ abuf,
    int k0, int nbase, int Kdim, int N, int tid) {
#if defined(HAVE_ASYNC_LDS)
  {  // W: 512 float4, 4 per thread — async, no VGPR data staging
    int f = tid;
#pragma unroll
    for (int i = 0; i < 4; ++i, f += 128) {
      const int kk = f >> 4;       // 16 float4 per 64-float row
      const int c4 = f & 15;
      GASYNC_B128(&Wb[(size_t)(k0 + kk) * N + nbase + c4 * 4],
                  &wbuf[kk * BN + c4 * 4]);
    }
  }
  {  // A: 256 float4, 2 per thread (padded rows, 144B-aligned)
    int f = tid;
#pragma unroll
    for (int i = 0; i < 2; ++i, f += 128) {
      const int row = f >> 3;      // 8 float4 per 32-float row
      const int c4 = f & 7;
      GASYNC_B128(&Ab[(size_t)row * ((size_t)E * Kdim) + k0 + c4 * 4],
                  &abuf[row * LDA + c4 * 4]);
    }
  }
#else
  {  // fallback: register staging
    int f = tid;
#pragma unroll
    for (int i = 0; i < 4; ++i, f += 128) {
      const int kk = f >> 4;
      const int c4 = f & 15;
      *(float4*)&wbuf[kk * BN + c4 * 4] =
          *(const float4*)&Wb[(size_t)(k0 + kk) * N + nbase + c4 * 4];
    }
    f = tid;
#pragma unroll
    for (int i = 0; i < 2; ++i, f += 128) {
      const int row = f >> 3;
      const int c4 = f & 7;
      *(float4*)&abuf[row * LDA + c4 * 4] =
          *(const float4*)&Ab[(size_t)row * ((size_t)E * Kdim) + k0 + c4 * 4];
    }
  }
#endif
}

template <bool RELU>
__global__ __launch_bounds__(128) void moe_gemm_kernel(
    const float* __restrict__ A, const float* __restrict__ W,
    const float* __restrict__ bias, float* __restrict__ C,
    int Kdim, int N) {
  const int e = blockIdx.y;
  const int nbase = blockIdx.x * BN;
  const int lane = threadIdx.x & 31;
  const int wave = threadIdx.x >> 5;  // 0..3
  const int tid = threadIdx.x;
  __shared__ __align__(16) float w_lds[2][KC * BN];   // 2 x 8 KB
  __shared__ __align__(16) float a_lds[2][32 * LDA];  // 2 x 4.5 KB

  const float* Ab = A + (size_t)e * Kdim;             // row m at Ab + m*E*Kdim
  const float* Wb = W + (size_t)e * Kdim * N;
  const int koff = (lane < 16) ? 0 : 2;
  const int nl = lane & 15;

  v8f acc0 = {0.f, 0.f, 0.f, 0.f, 0.f, 0.f, 0.f, 0.f};   // M tile 0..15
  v8f acc1 = {0.f, 0.f, 0.f, 0.f, 0.f, 0.f, 0.f, 0.f};   // M tile 16..31

  // prologue: start streaming chunk 0 into buffer 0
  stage_chunk(Wb, Ab, w_lds[0], a_lds[0], 0, nbase, Kdim, N, tid);

  int buf = 0;
  for (int k0 = 0; k0 < Kdim; k0 += KC) {
    wait_async_zero();     // my async loads for w_lds[buf]/a_lds[buf] done
    __syncthreads();       // everyone's loads done; prev compute on buf^1 done
    if (k0 + KC < Kdim)    // overlap: stream next chunk while computing this one
      stage_chunk(Wb, Ab, w_lds[buf ^ 1], a_lds[buf ^ 1], k0 + KC, nbase, Kdim, N, tid);

    const float* __restrict__ wl = w_lds[buf];
    const float* __restrict__ al = a_lds[buf];
#pragma unroll
    for (int kk = 0; kk < KC; kk += 4) {
      v2f b;
      b.x = wl[(kk + koff) * BN + wave * 16 + nl];
      b.y = wl[(kk + koff + 1) * BN + wave * 16 + nl];
      const float2 t0 = *(const float2*)&al[nl * LDA + kk + koff];
      const float2 t1 = *(const float2*)&al[(nl + 16) * LDA + kk + koff];
      v2f a0;  a0.x = t0.x;  a0.y = t0.y;
      v2f a1;  a1.x = t1.x;  a1.y = t1.y;
      acc0 = WMMA_F32_4(a0, b, acc0);
      acc1 = WMMA_F32_4(a1, b, acc1);
    }
    buf ^= 1;
  }

  // epilogue: bias (+relu), C[m, e, n]
  const int n = nbase + wave * 16 + nl;
  const float bv = bias[(size_t)e * N + n];
  const int mbase = (lane < 16) ? 0 : 8;
#pragma unroll
  for (int r = 0; r < 8; ++r) {
    float v0 = acc0[r] + bv;
    float v1 = acc1[r] + bv;
    if (RELU) { v0 = fmaxf(v0, 0.f); v1 = fmaxf(v1, 0.f); }
    const int m0 = mbase + r;
    const int m1 = mbase + r + 16;
    C[(size_t)m0 * E * N + (size_t)e * N + n] = v0;
    C[(size_t)m1 * E * N + (size_t)e * N + n] = v1;
  }
}

// =====================================================================
extern "C" void kernel_launch(void* const* d_in, const int* in_sizes, int n_in,
                              void* d_out, int out_size, void* d_ws, size_t ws_size,
                              hipStream_t stream) {
  const float* x  = (const float*)d_in[0];
  const float* Wg = (const float*)d_in[1];
  const float* bg = (const float*)d_in[2];
  const float* W1 = (const float*)d_in[3];
  const float* b1 = (const float*)d_in[4];
  const float* W2 = (const float*)d_in[5];
  const float* b2 = (const float*)d_in[6];
  const float* W3 = (const float*)d_in[7];
  const float* b3 = (const float*)d_in[8];
  float* out = (float*)d_out;

  char* ws = (char*)d_ws;
  size_t off = 0;
  float* logits = (float*)(ws + off);  off += (size_t)B * S * E * sizeof(float);     // 4 MB
  float* tkv    = (float*)(ws + off);  off += (size_t)B * E * KTOP * sizeof(float);
  int*   tki    = (int*)  (ws + off);  off += (size_t)B * E * KTOP * sizeof(int);
  off = (off + 255) & ~(size_t)255;
  float* inp    = (float*)(ws + off);  off += (size_t)B * E * KTOP * D * sizeof(float); // 8 MB
  float* h1     = (float*)(ws + off);  off += (size_t)B * E * O * sizeof(float);        // 2 MB
  float* h2     = (float*)(ws + off);  off += (size_t)B * E * O * sizeof(float);        // 2 MB

  gate_kernel<<<dim3((B * S) / 128), 256, 0, stream>>>(x, Wg, bg, logits);
  topk_kernel<<<dim3(B * E), 256, 0, stream>>>(logits, tki, tkv);
  gather_kernel<<<dim3(B * E), 256, 0, stream>>>(x, tki, tkv, inp);
  moe_gemm_kernel<true ><<<dim3(O / BN, E), 128, 0, stream>>>(inp, W1, b1, h1, KTOP * D, O);
  moe_gemm_kernel<true ><<<dim3(O / BN, E), 128, 0, stream>>>(h1,  W2, b2, h2, O, O);
  moe_gemm_kernel<false><<<dim3(O / BN, E), 128, 0, stream>>>(h2,  W3, b3, out, O, O);
}